// similar_RNN_31301721653612
// MI455X (gfx1250) — compile-verified
//
#include <hip/hip_runtime.h>
#include <hip/hip_bf16.h>
#include <math.h>
#include <stdint.h>

#define UNITS  1024
#define BATCH  64
#define TSTEPS 2048

typedef __attribute__((ext_vector_type(2))) float v2f;
typedef __attribute__((ext_vector_type(8))) float v8f;

// ---------------------------------------------------------------------------
// wave32 tree reduction (gfx1250 is wave32-only)
// ---------------------------------------------------------------------------
__device__ __forceinline__ float wave_reduce_add(float v) {
#pragma unroll
  for (int off = 16; off > 0; off >>= 1) v += __shfl_xor(v, off, 32);
  return v;
}

// ---------------------------------------------------------------------------
// Kernel 1: sequential RNN-like scan. One workgroup per batch element.
// 256 threads x 4 units each = 1024 units of state (s1, s2) in registers.
// Per step: two squared-diff reductions -> sigmoid(sqrt(mean)) -> blend.
// x is streamed (512 MB total); prefetch ahead to hide HBM latency.
// ---------------------------------------------------------------------------
__global__ __launch_bounds__(256) void rnn_scan_kernel(const float* __restrict__ x,
                                                       float* __restrict__ last) {
  const int b    = blockIdx.x;
  const int tid  = threadIdx.x;
  const int wave = tid >> 5;
  const int lane = tid & 31;

  __shared__ float red[2][8];

  float4 s1 = make_float4(0.f, 0.f, 0.f, 0.f);
  float4 s2 = make_float4(0.f, 0.f, 0.f, 0.f);

  const float4* xb = reinterpret_cast<const float4*>(x + (size_t)b * TSTEPS * UNITS) + tid;
  constexpr float inv_units = 1.0f / (float)UNITS;

  for (int t = 0; t < TSTEPS; ++t) {
    float4 xt = xb[(size_t)t * (UNITS / 4)];
    if (t + 8 < TSTEPS)
      __builtin_prefetch(&xb[(size_t)(t + 8) * (UNITS / 4)], 0, 0);  // global_prefetch_b8

    float a1 = 0.f, a2 = 0.f, dx;
    dx = xt.x - s1.x; a1 += dx * dx;
    dx = xt.y - s1.y; a1 += dx * dx;
    dx = xt.z - s1.z; a1 += dx * dx;
    dx = xt.w - s1.w; a1 += dx * dx;
    dx = xt.x - s2.x; a2 += dx * dx;
    dx = xt.y - s2.y; a2 += dx * dx;
    dx = xt.z - s2.z; a2 += dx * dx;
    dx = xt.w - s2.w; a2 += dx * dx;

    a1 = wave_reduce_add(a1);
    a2 = wave_reduce_add(a2);
    if (lane == 0) { red[0][wave] = a1; red[1][wave] = a2; }
    __syncthreads();
    float t1 = 0.f, t2 = 0.f;
#pragma unroll
    for (int w = 0; w < 8; ++w) { t1 += red[0][w]; t2 += red[1][w]; }
    __syncthreads();  // protect red[] before next iteration's writes

    float z1 = sqrtf(t1 * inv_units);
    float z2 = sqrtf(t2 * inv_units);
    float d1 = 1.0f / (1.0f + __expf(-z1));
    float d2 = 1.0f / (1.0f + __expf(-z2));

    // s2' blends the OLD s1 (per reference); keep a copy before overwriting.
    float4 o1 = s1;
    s1.x = d1 * o1.x + (1.f - d1) * xt.x;
    s1.y = d1 * o1.y + (1.f - d1) * xt.y;
    s1.z = d1 * o1.z + (1.f - d1) * xt.z;
    s1.w = d1 * o1.w + (1.f - d1) * xt.w;
    s2.x = (1.f - d2) * o1.x + d2 * xt.x;
    s2.y = (1.f - d2) * o1.y + d2 * xt.y;
    s2.z = (1.f - d2) * o1.z + d2 * xt.z;
    s2.w = (1.f - d2) * o1.w + d2 * xt.w;
  }

  float* L = last + (size_t)b * 2 * UNITS;  // last[b][0][:] = s1, last[b][1][:] = s2
  reinterpret_cast<float4*>(L)[tid]         = s1;
  reinterpret_cast<float4*>(L + UNITS)[tid] = s2;
}

// ---------------------------------------------------------------------------
// Kernel 2: QKV projections with fp32 WMMA + async B-panel staging.
//   Y = last(128x1024) @ W(1024x1024), W in {Wq, Wk, Wv}.
// Each block owns one 16-column N panel of one W. The full 1024x16 fp32
// B panel (64 KB) is copied global->LDS once per block with
// GLOBAL_LOAD_ASYNC_TO_LDS_B128 (ASYNCcnt), so the 8 waves (one 16x16
// output tile each, covering all 128 A rows) read B from LDS instead of
// re-reading global 8x. WMMA: V_WMMA_F32_16X16X4_F32, K-loop step 4.
//   A 16x4 layout: lane&15 = M row, (lane>>4)*2 = K pair base (v2f).
//   B 4x16 layout: lane&15 = N col, (lane>>4)*2 = K pair base.
//   C/D 16x16:     VGPR g -> row g + 8*(lane>>4), col = lane&15.
// LDS B layout Bs[k][c] (row stride 16 floats): half-wave reads of rows k
// and k+2 land in disjoint bank ranges -> conflict-free ds loads.
// ---------------------------------------------------------------------------
__global__ __launch_bounds__(256) void qkv_gemm_kernel(const float* __restrict__ last,
                                                       const float* __restrict__ Wq,
                                                       const float* __restrict__ Wk,
                                                       const float* __restrict__ Wv,
                                                       float* __restrict__ qkv) {
  const int nt   = blockIdx.x;          // N tile: 0..63
  const int mat  = blockIdx.y;          // which projection: 0..2
  const int wave = threadIdx.x >> 5;    // M tile: 0..7 (128 rows / 16)
  const int lane = threadIdx.x & 31;

  const float* W = (mat == 0) ? Wq : (mat == 1) ? Wk : Wv;
  float* out = qkv + (size_t)mat * (2 * BATCH) * UNITS;

  const int n0 = nt * 16;

  __shared__ float Bs[UNITS * 16];      // 64 KB B panel: Bs[k*16 + c]

  // ---- async global -> LDS staging of the B panel ----
  // 256 threads x 16 B each per pass (b128), 16 passes = 64 KB.
  {
    const int krow0 = threadIdx.x >> 2;        // 0..63
    const int c4    = (threadIdx.x & 3) << 2;  // 0,4,8,12
#pragma unroll
    for (int pass = 0; pass < 16; ++pass) {
      const int k = pass * 64 + krow0;
      const float* gp = W + (size_t)k * UNITS + n0 + c4;      // 16B aligned
      uint32_t lds_addr = (uint32_t)(uintptr_t)&Bs[k * 16 + c4];
      asm volatile("global_load_async_to_lds_b128 %0, %1, off"
                   :: "v"(lds_addr), "v"(gp)
                   : "memory");
    }
    asm volatile("s_wait_asynccnt 0x0" ::: "memory");
  }
  __syncthreads();

  const int m0  = wave * 16;
  const int row = lane & 15;            // A row / B col within tile
  const int kb  = (lane >> 4) << 1;     // 0 or 2: which K pair this lane holds

  const float* Arow = last + (size_t)(m0 + row) * UNITS;  // A[m0+row][*]

  v8f acc = {};
#pragma unroll 8
  for (int kk = 0; kk < UNITS; kk += 4) {
    v2f a, bm;
    a.x  = Arow[kk + kb];
    a.y  = Arow[kk + kb + 1];
    bm.x = Bs[(kk + kb) * 16 + row];
    bm.y = Bs[(kk + kb + 1) * 16 + row];
    // 8 args: (neg_a, A, neg_b, B, c_mod, C, reuse_a, reuse_b)
    acc = __builtin_amdgcn_wmma_f32_16x16x4_f32(false, a, false, bm,
                                                (short)0, acc, false, false);
  }

  const int orow = (lane >> 4) << 3;    // 0 or 8
  const int ocol = lane & 15;
#pragma unroll
  for (int g = 0; g < 8; ++g) {
    out[(size_t)(m0 + orow + g) * UNITS + n0 + ocol] = acc[g];
  }
}

// ---------------------------------------------------------------------------
// Kernel 3: 2x2 attention, only row 0 of the output is needed.
//   s0j = dot(q[b,0], k[b,j]) / 32 ; softmax over j ; out = w0*v0 + w1*v1.
// ---------------------------------------------------------------------------
__global__ __launch_bounds__(256) void attn_kernel(const float* __restrict__ qkv,
                                                   float* __restrict__ out) {
  const int b    = blockIdx.x;
  const int tid  = threadIdx.x;
  const int wave = tid >> 5;
  const int lane = tid & 31;

  const float* q = qkv;
  const float* k = qkv + (size_t)1 * (2 * BATCH) * UNITS;
  const float* v = qkv + (size_t)2 * (2 * BATCH) * UNITS;

  const float* q0 = q + (size_t)(2 * b) * UNITS;
  const float* k0 = k + (size_t)(2 * b) * UNITS;
  const float* k1 = k0 + UNITS;

  float p00 = 0.f, p01 = 0.f;
  for (int i = tid; i < UNITS; i += 256) {
    float qv = q0[i];
    p00 += qv * k0[i];
    p01 += qv * k1[i];
  }
  p00 = wave_reduce_add(p00);
  p01 = wave_reduce_add(p01);

  __shared__ float red[2][8];
  if (lane == 0) { red[0][wave] = p00; red[1][wave] = p01; }
  __syncthreads();
  float s00 = 0.f, s01 = 0.f;
#pragma unroll
  for (int w = 0; w < 8; ++w) { s00 += red[0][w]; s01 += red[1][w]; }

  const float scale = 1.0f / 32.0f;     // 1/sqrt(1024)
  s00 *= scale; s01 *= scale;
  float mx = fmaxf(s00, s01);
  float e0 = __expf(s00 - mx), e1 = __expf(s01 - mx);
  float inv = 1.0f / (e0 + e1);
  float w0 = e0 * inv, w1 = e1 * inv;

  const float* v0 = v + (size_t)(2 * b) * UNITS;
  const float* v1 = v0 + UNITS;
  for (int i = tid; i < UNITS; i += 256) {
    out[(size_t)b * UNITS + i] = w0 * v0[i] + w1 * v1[i];
  }
}

// ---------------------------------------------------------------------------
// Launch: scan -> WMMA projections -> attention. All on `stream`.
// Workspace layout (fp32): last[128*1024] | qkv[3*128*1024]  (2 MB total)
// ---------------------------------------------------------------------------
extern "C" void kernel_launch(void* const* d_in, const int* in_sizes, int n_in,
                              void* d_out, int out_size, void* d_ws, size_t ws_size,
                              hipStream_t stream) {
  const float* x  = (const float*)d_in[0];
  const float* Wq = (const float*)d_in[1];
  const float* Wk = (const float*)d_in[2];
  const float* Wv = (const float*)d_in[3];

  float* ws   = (float*)d_ws;
  float* last = ws;                                  // 128*1024
  float* qkv  = ws + (size_t)(2 * BATCH) * UNITS;    // 3 * 128*1024

  rnn_scan_kernel<<<dim3(BATCH), dim3(256), 0, stream>>>(x, last);
  qkv_gemm_kernel<<<dim3(UNITS / 16, 3), dim3(256), 0, stream>>>(last, Wq, Wk, Wv, qkv);
  attn_kernel<<<dim3(BATCH), dim3(256), 0, stream>>>(qkv, (float*)d_out);
}